// SRURNNGate_52956946759996
// MI455X (gfx1250) — compile-verified
//
#include <hip/hip_runtime.h>
#include <stdint.h>

// Problem sizes (match reference)
#define Bsz 16
#define Lsz 2048
#define Dsz 1024

#define TPB    64                 // threads per block (2 wave32s) — waves are independent
#define WPB    (TPB / 32)         // 2 waves per block
#define DSLICE TPB                // channels per block
#define KDEPTH 32                 // ring depth in chunks per wave (2K=64 async instrs in flight)
#define SPC    4                  // timesteps per chunk
#define NCHUNK (Lsz / SPC)        // 512 chunks
// per-wave LDS: f ring KDEPTH*512B + u ring KDEPTH*512B = 32 KB
#define WAVE_LDS_FLOATS (KDEPTH * 256)
#define BLOCK_LDS_BYTES (WPB * KDEPTH * 1024)   // 65536 B

__device__ __forceinline__ float sru_step(float c, float fv, float uv, float w) {
    // g = sigmoid(fv + w*c) ; c' = (1-g)*c + g*uv = c + g*(uv - c)
    float x = fv + w * c;
    float e = __builtin_amdgcn_exp2f(-1.4426950408889634f * x);  // exp(-x)
    float g = __builtin_amdgcn_rcpf(1.0f + e);
    return c + g * (uv - c);
}

__global__ void __launch_bounds__(TPB, 1)
sru_scan_kernel(const float* __restrict__ f, const float* __restrict__ u,
                const float* __restrict__ Wfc, float* __restrict__ out_cs,
                float* __restrict__ out_clast)
{
    extern __shared__ float smem[];

    const int tid  = (int)threadIdx.x;
    const int lane = tid & 31;
    const int wv   = tid >> 5;
    const int blk  = (int)blockIdx.x;            // 256 blocks
    const int b    = blk >> 4;                   // batch (16)
    const int dblk = (blk & 15) * DSLICE;        // 64-wide channel slice
    const int d0   = dblk + wv * 32;             // wave's first channel
    const int d    = d0 + lane;                  // this thread's channel

    // Uniform 64-bit bases for GVS-mode async loads (SGPR pair)
    const uint64_t fbase = (uint64_t)(uintptr_t)(f + (size_t)b * Lsz * Dsz);
    const uint64_t ubase = (uint64_t)(uintptr_t)(u + (size_t)b * Lsz * Dsz);

    // Per-lane global byte offset within a chunk:
    // lane i covers timestep (i>>3) of the chunk, channels d0+(i&7)*4 .. +3 (16B)
    const uint32_t voff0 =
        (uint32_t)(d0 + (lane & 7) * 4) * 4u + (uint32_t)(lane >> 3) * (Dsz * 4u);

    // LDS byte addresses (low 32 bits of flat pointer == LDS offset on CDNA5)
    const uint32_t lds_base = (uint32_t)(uintptr_t)smem;
    const uint32_t wv_base  = lds_base + (uint32_t)wv * (KDEPTH * 1024u);
    const uint32_t ldsf0 = wv_base + (uint32_t)lane * 16u;                 // + slot*512
    const uint32_t ldsu0 = wv_base + KDEPTH * 512u + (uint32_t)lane * 16u; // + slot*512

    // Float-index bases for LDS consumption: smem[fi + slot*128 + q*32]
    const int fbaseIdx = wv * WAVE_LDS_FLOATS + lane;
    const int ubaseIdx = fbaseIdx + KDEPTH * 128;

    const float w = Wfc[d];
    float c = 0.0f;
    float* outp = out_cs + (size_t)b * Lsz * Dsz + d;

    // ---- prologue: prefetch chunks 0..KDEPTH-1 into the ring ----
    // (issue of the 64th async op may briefly stall at the 6-bit ASYNCcnt —
    //  equivalent to the wait we'd perform anyway; harmless)
    for (int j = 0; j < KDEPTH; ++j) {
        uint32_t go = voff0 + (uint32_t)j * (SPC * Dsz * 4u);
        uint32_t lf = ldsf0 + (uint32_t)j * 512u;
        uint32_t lu = ldsu0 + (uint32_t)j * 512u;
        asm volatile("global_load_async_to_lds_b128 %0, %1, %2"
                     :: "v"(lf), "v"(go), "s"(fbase) : "memory");
        asm volatile("global_load_async_to_lds_b128 %0, %1, %2"
                     :: "v"(lu), "v"(go), "s"(ubase) : "memory");
    }

#define CONSUME_CHUNK(JJ)                                                      \
    do {                                                                       \
        const int s_  = (JJ) & (KDEPTH - 1);                                   \
        const int fi_ = fbaseIdx + s_ * 128;                                   \
        const int ui_ = ubaseIdx + s_ * 128;                                   \
        float f0 = smem[fi_ +  0], f1 = smem[fi_ + 32];                        \
        float f2 = smem[fi_ + 64], f3 = smem[fi_ + 96];                        \
        float u0 = smem[ui_ +  0], u1 = smem[ui_ + 32];                        \
        float u2 = smem[ui_ + 64], u3 = smem[ui_ + 96];                        \
        c = sru_step(c, f0, u0, w);                                            \
        __builtin_nontemporal_store(c, &outp[0 * Dsz]);                        \
        c = sru_step(c, f1, u1, w);                                            \
        __builtin_nontemporal_store(c, &outp[1 * Dsz]);                        \
        c = sru_step(c, f2, u2, w);                                            \
        __builtin_nontemporal_store(c, &outp[2 * Dsz]);                        \
        c = sru_step(c, f3, u3, w);                                            \
        __builtin_nontemporal_store(c, &outp[3 * Dsz]);                        \
        outp += SPC * Dsz;                                                     \
    } while (0)

    // ---- steady state: wait oldest chunk, consume, refill same slot ----
    for (int j = 0; j < NCHUNK - KDEPTH; ++j) {
        // 2 async instrs per chunk; <= 2*(KDEPTH-1)=62 outstanding => chunk j landed
        asm volatile("s_wait_asynccnt 62" ::: "memory");
        CONSUME_CHUNK(j);
        // our ds_loads from this slot must retire before the async engine
        // overwrites it (DScnt and ASYNCcnt are unordered w.r.t. each other);
        // dscnt is almost certainly already 0 here since compute consumed the data
        asm volatile("s_wait_dscnt 0" ::: "memory");
        {
            const int jp = j + KDEPTH;                // same ring slot as j
            uint32_t go = voff0 + (uint32_t)jp * (SPC * Dsz * 4u);
            uint32_t lf = ldsf0 + (uint32_t)(j & (KDEPTH - 1)) * 512u;
            uint32_t lu = ldsu0 + (uint32_t)(j & (KDEPTH - 1)) * 512u;
            asm volatile("global_load_async_to_lds_b128 %0, %1, %2"
                         :: "v"(lf), "v"(go), "s"(fbase) : "memory");
            asm volatile("global_load_async_to_lds_b128 %0, %1, %2"
                         :: "v"(lu), "v"(go), "s"(ubase) : "memory");
        }
    }

    // ---- tail: everything already issued; drain ----
    asm volatile("s_wait_asynccnt 0" ::: "memory");
    for (int j = NCHUNK - KDEPTH; j < NCHUNK; ++j) {
        CONSUME_CHUNK(j);
    }
#undef CONSUME_CHUNK

    out_clast[(size_t)b * Dsz + d] = c;
}

extern "C" void kernel_launch(void* const* d_in, const int* in_sizes, int n_in,
                              void* d_out, int out_size, void* d_ws, size_t ws_size,
                              hipStream_t stream) {
    (void)in_sizes; (void)n_in; (void)out_size; (void)d_ws; (void)ws_size;
    const float* f   = (const float*)d_in[0];
    const float* u   = (const float*)d_in[1];
    const float* Wfc = (const float*)d_in[2];
    float* out_cs    = (float*)d_out;
    float* out_clast = out_cs + (size_t)Bsz * Lsz * Dsz;

    dim3 grid(Bsz * (Dsz / DSLICE));   // 16 batches * 16 channel slices = 256 blocks
    dim3 block(TPB);
    sru_scan_kernel<<<grid, block, BLOCK_LDS_BYTES, stream>>>(f, u, Wfc, out_cs, out_clast);
}